// QPSolver_51075751084509
// MI455X (gfx1250) — compile-verified
//
#include <hip/hip_runtime.h>

// ---------------------------------------------------------------------------
// Batched QP PDHG solver for MI455X (gfx1250, wave32, WMMA).
// One workgroup (256 threads = 8 waves) per batch item; all matrices in LDS.
// Small GEMMs use V_WMMA_F32_16X16X4_F32 (f32 needed: matrix inversion).
// ---------------------------------------------------------------------------

#define BSZ      4096
#define NDIM     48          // n
#define MCON     64          // m
#define ALPHA    1.0f
#define BETA     1.0f
#define NTHREADS 256
#define NWAVES   (NTHREADS / 32)
#define AUGLD    96          // stride of augmented Gauss-Jordan buffer

typedef float v2f __attribute__((ext_vector_type(2)));
typedef float v8f __attribute__((ext_vector_type(8)));

// ---- WMMA f32 16x16x4 fragment helpers (CDNA5 ISA 7.12.2 layouts) ----------
// A (16x4): lanes 0-15 -> M=0..15 K={k0,k0+1}; lanes 16-31 -> same M, K+2.
__device__ __forceinline__ v2f ld_frag_a(const float* A, int rs, int cs,
                                         int m0, int k0, int lane) {
  int m  = m0 + (lane & 15);
  int kk = k0 + 2 * (lane >> 4);
  v2f a;
  a.x = A[m * rs + kk * cs];
  a.y = A[m * rs + (kk + 1) * cs];
  return a;
}
// B (4x16): lanes 0-15 -> N=0..15 K={k0,k0+1}; lanes 16-31 -> same N, K+2.
__device__ __forceinline__ v2f ld_frag_b(const float* B, int rs, int cs,
                                         int k0, int n0, int lane) {
  int n  = n0 + (lane & 15);
  int kk = k0 + 2 * (lane >> 4);
  v2f b;
  b.x = B[kk * rs + n * cs];
  b.y = B[(kk + 1) * rs + n * cs];
  return b;
}
// C/D (16x16 f32, 8 VGPRs): lane half selects M base (0 or 8), VGPR v -> M+v.
__device__ __forceinline__ void st_frag_d(float* D, int ldd, int m0, int n0,
                                          int lane, v8f c) {
  int n  = n0 + (lane & 15);
  int mh = m0 + 8 * (lane >> 4);
#pragma unroll
  for (int v = 0; v < 8; ++v) D[(mh + v) * ldd + n] = c[v];
}

// Generic small GEMM D = A*B via WMMA; strides (rs,cs) allow transposed reads.
// Tiles distributed across the 8 waves; loop bounds are wave-uniform so EXEC
// stays all-ones around every v_wmma (ISA 7.12 requirement).
__device__ void wmma_gemm_f32(const float* A, int ars, int acs,
                              const float* B, int brs, int bcs,
                              float* D, int ldd,
                              int M, int N, int K, int wave, int lane) {
  const int NT = N >> 4;
  const int ntiles = (M >> 4) * NT;
  for (int t = wave; t < ntiles; t += NWAVES) {
    const int tm = (t / NT) << 4;
    const int tn = (t % NT) << 4;
    v8f c = {0.f, 0.f, 0.f, 0.f, 0.f, 0.f, 0.f, 0.f};
    for (int k = 0; k < K; k += 4) {
      v2f a = ld_frag_a(A, ars, acs, tm, k, lane);
      v2f b = ld_frag_b(B, brs, bcs, k, tn, lane);
      c = __builtin_amdgcn_wmma_f32_16x16x4_f32(false, a, false, b,
                                                (short)0, c, false, false);
    }
    st_frag_d(D, ldd, tm, tn, lane, c);
  }
}

// Gauss-Jordan on an NDIM x ncols augmented system in LDS (stride AUGLD).
// No pivoting (matrices are SPD). Column snapshot avoids intra-step races.
__device__ void gauss_jordan(float* aug, int ncols, float* colbuf, int tid) {
  for (int p = 0; p < NDIM; ++p) {
    for (int r = tid; r < NDIM; r += NTHREADS) colbuf[r] = aug[r * AUGLD + p];
    __syncthreads();
    const float pivinv = 1.0f / colbuf[p];
    for (int c = tid; c < ncols; c += NTHREADS) aug[p * AUGLD + c] *= pivinv;
    __syncthreads();
    for (int e = tid; e < NDIM * ncols; e += NTHREADS) {
      int r = e / ncols;
      int c = e - r * ncols;
      if (r != p)
        aug[r * AUGLD + c] = fmaf(-colbuf[r], aug[p * AUGLD + c],
                                  aug[r * AUGLD + c]);
    }
    __syncthreads();
  }
}

__global__ __launch_bounds__(NTHREADS) void qp_pdhg_kernel(
    const float* __restrict__ q, const float* __restrict__ bvec,
    const float* __restrict__ P, const float* __restrict__ H,
    const float* __restrict__ X0, const int* __restrict__ iters_p,
    float* __restrict__ out) {
  __shared__ float sH[MCON * NDIM];    // H, 64x48 row-major
  __shared__ float sAug[NDIM * AUGLD]; // 48x96 Gauss-Jordan workspace
  __shared__ float sT[MCON * NDIM];    // holds HtH (48x48) early, then H*Ginv
  __shared__ float sTD[MCON * MCON];   // tD, 64x64
  __shared__ float sB[MCON], sX[MCON], sZ[MCON], sS[MCON], sW[MCON], sMu[MCON];
  __shared__ float sPq[NDIM], sCol[NDIM];
  __shared__ float sRed[NTHREADS];

  const int tid  = threadIdx.x;
  const int lane = tid & 31;
  const int wave = tid >> 5;
  const int bidx = blockIdx.x;

  const float* Hb = H + (size_t)bidx * MCON * NDIM;
  const float* Pb = P + (size_t)bidx * NDIM * NDIM;

  // ---- Stage 0: stage inputs into LDS --------------------------------------
  for (int e = tid; e < MCON * NDIM; e += NTHREADS) sH[e] = Hb[e];
  if (tid < MCON) {
    sB[tid] = bvec[bidx * MCON + tid];
    sX[tid] = X0[bidx * 2 * MCON + tid];
    sZ[tid] = X0[bidx * 2 * MCON + MCON + tid];
  }
  for (int e = tid; e < NDIM * NDIM; e += NTHREADS) {
    int r = e / NDIM, c = e - r * NDIM;
    sAug[r * AUGLD + c] = Pb[e];               // [P | q] for the P-solve
  }
  if (tid < NDIM) sAug[tid * AUGLD + NDIM] = q[bidx * NDIM + tid];
  __syncthreads();

  // ---- Stage 1: HtH = H^T * H (48x48) via WMMA, into sT --------------------
  // A = H^T -> element (m,k) = sH[k*NDIM+m]; B = H.
  wmma_gemm_f32(sH, 1, NDIM, sH, NDIM, 1, sT, NDIM,
                NDIM, NDIM, MCON, wave, lane);
  __syncthreads();

  // ---- Stage 2: solve P * Pq = q (Gauss-Jordan on [P|q]) -------------------
  gauss_jordan(sAug, NDIM + 1, sCol, tid);
  if (tid < NDIM) sPq[tid] = sAug[tid * AUGLD + NDIM];
  __syncthreads();

  // ---- Stage 3: build [G | I], G = P + HtH; invert -> Ginv in cols 48..95 --
  for (int e = tid; e < NDIM * NDIM; e += NTHREADS) {
    int r = e / NDIM, c = e - r * NDIM;
    sAug[r * AUGLD + c]        = Pb[e] + sT[e];       // sT holds HtH (ld=48)
    sAug[r * AUGLD + NDIM + c] = (r == c) ? 1.0f : 0.0f;
  }
  __syncthreads();
  gauss_jordan(sAug, 2 * NDIM, sCol, tid);

  // ---- Stage 4: T = H * Ginv (64x48) via WMMA (overwrites sT) --------------
  wmma_gemm_f32(sH, NDIM, 1, &sAug[NDIM], AUGLD, 1, sT, NDIM,
                MCON, NDIM, NDIM, wave, lane);
  __syncthreads();

  // ---- Stage 5: M2 = T * H^T (64x64) via WMMA; tD = I/beta - M2 ------------
  wmma_gemm_f32(sT, NDIM, 1, sH, 1, NDIM, sTD, MCON,
                MCON, MCON, NDIM, wave, lane);
  __syncthreads();
  for (int e = tid; e < MCON * MCON; e += NTHREADS) {
    int r = e >> 6, c = e & 63;
    sTD[e] = ((r == c) ? (1.0f / BETA) : 0.0f) - sTD[e];
  }
  __syncthreads();

  // ---- Stage 6: mu = tD * (H*Pq - b) ---------------------------------------
  if (tid < MCON) {
    float acc = 0.f;
#pragma unroll 8
    for (int j = 0; j < NDIM; ++j) acc = fmaf(sH[tid * NDIM + j], sPq[j], acc);
    sW[tid] = acc - sB[tid];
  }
  __syncthreads();
  if (tid < MCON) {
    float acc = 0.f;
#pragma unroll 8
    for (int j = 0; j < MCON; ++j) acc = fmaf(sTD[tid * MCON + j], sW[j], acc);
    sMu[tid] = acc;
    sS[tid]  = sX[tid] * (1.0f / BETA) + sZ[tid];  // s = x/beta + z
  }
  __syncthreads();

  // ---- Stage 7: PDHG iterations: w = tD*s + mu; x=w; z=relu(x+z-2*alpha*w) -
  const int niters = iters_p[0];
  for (int it = 0; it < niters; ++it) {
    {  // 4 lanes per row, 16 MACs each, reduce in LDS
      const int row = tid >> 2, part = tid & 3;
      const float* tr = &sTD[row * MCON + part * 16];
      const float* sv = &sS[part * 16];
      float acc = 0.f;
#pragma unroll
      for (int j = 0; j < 16; ++j) acc = fmaf(tr[j], sv[j], acc);
      sRed[tid] = acc;
    }
    __syncthreads();
    if (tid < MCON) {
      const float w = sRed[4 * tid] + sRed[4 * tid + 1] +
                      sRed[4 * tid + 2] + sRed[4 * tid + 3] + sMu[tid];
      const float told = sX[tid] + sZ[tid];
      const float zn = fmaxf(fmaf(-2.0f * ALPHA, w, told), 0.0f);
      sX[tid] = w;
      sZ[tid] = zn;
      sS[tid] = fmaf(w, 1.0f / BETA, zn);
    }
    __syncthreads();
  }

  // ---- Stage 8: write X = [x ; z] ------------------------------------------
  if (tid < 2 * MCON)
    out[(size_t)bidx * 2 * MCON + tid] = (tid < MCON) ? sX[tid] : sZ[tid - MCON];

  // ---- Stage 9: primal = (HtH)^-1 * H^T (z - b) ----------------------------
  // Recompute HtH via WMMA straight into the augmented buffer (cols 0..47),
  // put rhs = H^T(z-b) into col 48, solve.
  wmma_gemm_f32(sH, 1, NDIM, sH, NDIM, 1, sAug, AUGLD,
                NDIM, NDIM, MCON, wave, lane);
  if (tid < NDIM) {
    float acc = 0.f;
#pragma unroll 8
    for (int j = 0; j < MCON; ++j)
      acc = fmaf(sH[j * NDIM + tid], sZ[j] - sB[j], acc);
    sAug[tid * AUGLD + NDIM] = acc;
  }
  __syncthreads();
  gauss_jordan(sAug, NDIM + 1, sCol, tid);
  if (tid < NDIM)
    out[(size_t)BSZ * 2 * MCON + (size_t)bidx * NDIM + tid] =
        sAug[tid * AUGLD + NDIM];
}

extern "C" void kernel_launch(void* const* d_in, const int* in_sizes, int n_in,
                              void* d_out, int out_size, void* d_ws,
                              size_t ws_size, hipStream_t stream) {
  const float* q     = (const float*)d_in[0];
  const float* b     = (const float*)d_in[1];
  const float* P     = (const float*)d_in[2];
  const float* H     = (const float*)d_in[3];
  const float* X0    = (const float*)d_in[4];
  const int*   iters = (const int*)d_in[5];
  float* out = (float*)d_out;
  (void)in_sizes; (void)n_in; (void)out_size; (void)d_ws; (void)ws_size;
  qp_pdhg_kernel<<<BSZ, NTHREADS, 0, stream>>>(q, b, P, H, X0, iters, out);
}